// BootstrappedBCEWithLogitsLoss_57509612093595
// MI455X (gfx1250) — compile-verified
//
#include <hip/hip_runtime.h>
#include <hip/hip_bf16.h>
#include <stdint.h>

// ---------------- problem constants ----------------
#define BATCH   64
#define HW      (512 * 512)          // 262144 pixels per row
#define K_TOP   65536                // 0.25 * HW
#define TOTALK  (BATCH * K_TOP)      // 4194304 selected elements

// histogram select parameters
#define NBINS    4096
#define NBINS2   4096
#define LOSS_MAX 8.0f
#define SCALE1   ((float)NBINS / LOSS_MAX)   // 512 bins per unit loss

// streaming tiles: 8 tiles of 2048 f32 per block, double-buffered TDM
#define TILE    2048
#define TPB     8                    // tiles per block
#define CHUNK   (TILE * TPB)         // 16384 elements per block
#define SLICES  (HW / CHUNK)         // 16 slices per row

// ---------------- workspace layout (32-bit words) ----------------
#define W_H1C  0                          // [BATCH*NBINS]  u32 counts, level 1
#define W_H1S  (W_H1C + BATCH * NBINS)    // [BATCH*NBINS]  f32 sums,   level 1
#define W_INFO (W_H1S + BATCH * NBINS)    // [BATCH*4]      {t1, rem, sum_above, pad}
#define W_H2C  (W_INFO + BATCH * 4)       // [BATCH*NBINS2] u32 counts, level 2
#define W_H2S  (W_H2C + BATCH * NBINS2)   // [BATCH*NBINS2] f32 sums,   level 2
#define W_END  (W_H2S + BATCH * NBINS2)   // ~4.2 MB total

// ---------------- CDNA5 feature detection ----------------
typedef uint32_t u32x4_t __attribute__((ext_vector_type(4)));
typedef int      i32x8_t __attribute__((ext_vector_type(8)));
typedef int      i32x4_t __attribute__((ext_vector_type(4)));
typedef float    v2f     __attribute__((ext_vector_type(2)));
typedef float    v8f     __attribute__((ext_vector_type(8)));

#if __has_builtin(__builtin_amdgcn_tensor_load_to_lds) && __has_builtin(__builtin_amdgcn_s_wait_tensorcnt)
#define HAVE_TDM 1
#else
#define HAVE_TDM 0
#endif

#if __has_builtin(__builtin_amdgcn_wmma_f32_16x16x4_f32)
#define HAVE_WMMA_F32 1
#else
#define HAVE_WMMA_F32 0
#endif

// ---------------- device helpers ----------------
__device__ __forceinline__ float bce_loss(float x, float y) {
    // max(x,0) - x*y + log1p(exp(-|x|))  (>= 0 always)
    return fmaxf(x, 0.0f) - x * y + log1pf(expf(-fabsf(x)));
}

#if HAVE_TDM
// 1D contiguous TDM load: n f32 elements from gsrc -> LDS byte offset lds_off.
// D# built per CDNA5 ISA (groups 0/1; groups 2/3 unused -> zero).
// 6-arg builtin: (uint32x4 g0, int32x8 g1, int32x4 g2, int32x4 g3, int32x8, i32 cpol)
__device__ __forceinline__ void tdm_load_f32_1d(uint32_t lds_off, const float* gsrc, uint32_t n) {
    uint64_t ga = (uint64_t)(uintptr_t)gsrc;
    u32x4_t g0;
    g0[0] = 1u;                                            // count=1, is_restore=0, gather=0
    g0[1] = lds_off;                                       // lds_addr (bytes)
    g0[2] = (uint32_t)ga;                                  // global_addr[31:0]
    g0[3] = (uint32_t)((ga >> 32) & 0x01FFFFFFu)           // global_addr[56:32]
          | (2u << 30);                                    // type = 2 ("image")
    i32x8_t g1;
    g1[0] = (int)(2u << 16);                               // workgroup_mask=0, data_size=2 (4B)
    g1[1] = (int)((n & 0xFFFFu) << 16);                    // tensor_dim0[15:0] (bits 63:48)
    g1[2] = (int)(((n >> 16) & 0xFFFFu) | (1u << 16));     // tensor_dim0[31:16] | tensor_dim1=1
    g1[3] = (int)((n & 0xFFFFu) << 16);                    // tensor_dim1 hi=0 | tile_dim0=n
    g1[4] = 0;                                             // tile_dim1=0, tile_dim2=0
    g1[5] = (int)n;                                        // tensor_dim0_stride[31:0]
    g1[6] = 0;                                             // stride0 hi | tensor_dim1_stride lo
    g1[7] = 0;                                             // tensor_dim1_stride hi
    i32x4_t z4 = {0, 0, 0, 0};
    i32x8_t z8 = {0, 0, 0, 0, 0, 0, 0, 0};
    __builtin_amdgcn_tensor_load_to_lds(g0, g1, z4, z4, z8, 0);
}
#endif

// Exact-f32 wave32 reduction of one value per lane, via V_WMMA_F32_16X16X4_F32.
// A = 16x4 f32 tile (2 VGPRs/lane), B = ones(4x16) => D[m][n] = v[m] + v[m+16].
// Per-lane sum of the 8 D VGPRs gives rows 0..7 (lanes 0-15) / 8..15 (lanes 16-31);
// one shfl_xor(16) completes the total (broadcast to all lanes).
__device__ __forceinline__ float wave_reduce_sum(float v) {
#if HAVE_WMMA_F32
    v2f a; a.x = v;    a.y = 0.0f;
    v2f b; b.x = 1.0f; b.y = 1.0f;
    v8f c = {};
    v8f d = __builtin_amdgcn_wmma_f32_16x16x4_f32(false, a, false, b,
                                                  (short)0, c, false, false);
    float s = d[0] + d[1] + d[2] + d[3] + d[4] + d[5] + d[6] + d[7];
    s += __shfl_xor(s, 16, 32);
    return s;
#else
    for (int off = 16; off > 0; off >>= 1) v += __shfl_down(v, off, 32);
    return v;   // valid in lane 0
#endif
}

// ---------------- kernels ----------------

__global__ void btk_init(uint32_t* __restrict__ ws) {
    size_t i = (size_t)blockIdx.x * 256 + threadIdx.x;
    if (i < (size_t)W_END) ws[i] = 0u;
}

// Pass 1: level-1 histogram (count + f32 sum) per row.
// Double-buffered TDM pipeline: DMA of tile t+1 overlaps VALU/LDS work on tile t
// (TENSORcnt is in-order per wave, so s_wait_tensorcnt(2) retires exactly the
// current tile's two loads while the next tile's two stay in flight).
__global__ __launch_bounds__(256) void btk_hist1(const float* __restrict__ pred,
                                                 const float* __restrict__ gts,
                                                 uint32_t* __restrict__ ws) {
    __shared__ float        ldsP[2][TILE];
    __shared__ float        ldsG[2][TILE];
    __shared__ unsigned int lcnt[NBINS];
    __shared__ float        lsum[NBINS];

    const int row = blockIdx.y;
    const int tid = threadIdx.x;
    const size_t base = (size_t)row * HW + (size_t)blockIdx.x * CHUNK;

#if HAVE_TDM
    if (tid < 32) {  // one wave drives the DMA; TDM ignores EXEC, TENSORcnt is per-wave
        tdm_load_f32_1d((uint32_t)(uintptr_t)&ldsP[0][0], pred + base, TILE);
        tdm_load_f32_1d((uint32_t)(uintptr_t)&ldsG[0][0], gts  + base, TILE);
    }
#endif
    for (int i = tid; i < NBINS; i += 256) { lcnt[i] = 0u; lsum[i] = 0.0f; }

    for (int t = 0; t < TPB; ++t) {
        const int cur = t & 1;
#if HAVE_TDM
        if (tid < 32) {
            if (t + 1 < TPB) {   // prefetch next tile into the other buffer
                const size_t nb = base + (size_t)(t + 1) * TILE;
                tdm_load_f32_1d((uint32_t)(uintptr_t)&ldsP[cur ^ 1][0], pred + nb, TILE);
                tdm_load_f32_1d((uint32_t)(uintptr_t)&ldsG[cur ^ 1][0], gts  + nb, TILE);
                __builtin_amdgcn_s_wait_tensorcnt(2);  // current tile's 2 loads done
            } else {
                __builtin_amdgcn_s_wait_tensorcnt(0);  // drain for last tile
            }
        }
        __syncthreads();   // data visible to all waves (also guards hist zeroing at t==0)
#else
        for (int i = tid; i < TILE; i += 256) {
            ldsP[cur][i] = pred[base + (size_t)t * TILE + i];
            ldsG[cur][i] = gts [base + (size_t)t * TILE + i];
        }
        __syncthreads();
#endif
        for (int i = tid; i < TILE; i += 256) {
            float loss = bce_loss(ldsP[cur][i], ldsG[cur][i]);
            int bin = (int)(loss * SCALE1);
            bin = bin < 0 ? 0 : (bin >= NBINS ? NBINS - 1 : bin);
            atomicAdd(&lcnt[bin], 1u);      // ds_add_u32
            atomicAdd(&lsum[bin], loss);    // ds_add_f32
        }
        __syncthreads();   // all reads of buf[cur] done before it is refilled at t+2
    }

    unsigned int* gcnt = (unsigned int*)(ws + W_H1C) + (size_t)row * NBINS;
    float*        gsum = (float*)(ws + W_H1S) + (size_t)row * NBINS;
    for (int i = tid; i < NBINS; i += 256) {
        unsigned int c = lcnt[i];
        if (c) { atomicAdd(&gcnt[i], c); atomicAdd(&gsum[i], lsum[i]); }
    }
}

// Find per-row threshold bin t1, remaining count, and exact sum of all bins above t1.
__global__ void btk_select(uint32_t* __restrict__ ws) {
    int row = threadIdx.x;
    if (row >= BATCH) return;
    const unsigned int* cnt = (const unsigned int*)(ws + W_H1C) + (size_t)row * NBINS;
    const float*        sum = (const float*)(ws + W_H1S) + (size_t)row * NBINS;
    unsigned int cum = 0; float s = 0.0f; int t1 = 0; unsigned int rem = K_TOP;
    for (int b = NBINS - 1; b >= 0; --b) {
        unsigned int c = cnt[b];
        if (cum + c >= (unsigned int)K_TOP) { t1 = b; rem = K_TOP - cum; break; }
        cum += c; s += sum[b];
    }
    uint32_t* info = ws + W_INFO + (size_t)row * 4;
    info[0] = (uint32_t)t1;
    info[1] = rem;
    ((float*)info)[2] = s;
}

// Pass 2: refine the threshold bin into NBINS2 sub-bins (count + sum of full loss).
// float4 streaming (global_load_b128): most elements miss the t1 filter, so this
// pass is pure bandwidth + a handful of LDS atomics.
__global__ __launch_bounds__(256) void btk_hist2(const float* __restrict__ pred,
                                                 const float* __restrict__ gts,
                                                 uint32_t* __restrict__ ws) {
    __shared__ unsigned int lcnt[NBINS2];
    __shared__ float        lsum[NBINS2];

    const int row = blockIdx.y;
    const int tid = threadIdx.x;
    const uint32_t t1 = (ws + W_INFO + (size_t)row * 4)[0];
    const float lo = (float)t1 * (LOSS_MAX / (float)NBINS);

    for (int i = tid; i < NBINS2; i += 256) { lcnt[i] = 0u; lsum[i] = 0.0f; }
    __syncthreads();

    const size_t base = (size_t)row * HW + (size_t)blockIdx.x * CHUNK;
    const float4* p4 = (const float4*)(pred + base);
    const float4* g4 = (const float4*)(gts  + base);
    for (int v = tid; v < CHUNK / 4; v += 256) {
        float4 xp = p4[v];
        float4 yg = g4[v];
        const float xs[4] = {xp.x, xp.y, xp.z, xp.w};
        const float ys[4] = {yg.x, yg.y, yg.z, yg.w};
        #pragma unroll
        for (int j = 0; j < 4; ++j) {
            float loss = bce_loss(xs[j], ys[j]);
            int bin = (int)(loss * SCALE1);
            bin = bin < 0 ? 0 : (bin >= NBINS ? NBINS - 1 : bin);
            if ((uint32_t)bin == t1) {
                float frac = (loss - lo) * SCALE1;       // [0,1) within the bin
                int sb = (int)(frac * (float)NBINS2);
                sb = sb < 0 ? 0 : (sb >= NBINS2 ? NBINS2 - 1 : sb);
                atomicAdd(&lcnt[sb], 1u);
                atomicAdd(&lsum[sb], loss);
            }
        }
    }
    __syncthreads();

    unsigned int* gc = (unsigned int*)(ws + W_H2C) + (size_t)row * NBINS2;
    float*        gs = (float*)(ws + W_H2S) + (size_t)row * NBINS2;
    for (int i = tid; i < NBINS2; i += 256) {
        unsigned int c = lcnt[i];
        if (c) { atomicAdd(&gc[i], c); atomicAdd(&gs[i], lsum[i]); }
    }
}

// Per-row top-k sum -> WMMA cross-lane reduction -> scalar mean.
__global__ void btk_finalize(uint32_t* __restrict__ ws, float* __restrict__ out) {
    __shared__ float wpart[2];
    const int t = threadIdx.x;

    float rowtop = 0.0f;
    if (t < BATCH) {
        const uint32_t* info = ws + W_INFO + (size_t)t * 4;
        const unsigned int rem = info[1];
        const float s_above = ((const float*)info)[2];
        const unsigned int* cnt = (const unsigned int*)(ws + W_H2C) + (size_t)t * NBINS2;
        const float*        sum = (const float*)(ws + W_H2S) + (size_t)t * NBINS2;
        unsigned int cum = 0; float s2 = 0.0f;
        for (int b = NBINS2 - 1; b >= 0; --b) {
            unsigned int c = cnt[b];
            if (cum + c >= rem) {
                if (c > 0 && rem > cum)                      // ties in final sub-bin: average
                    s2 += (float)(rem - cum) * (sum[b] / (float)c);
                break;
            }
            cum += c; s2 += sum[b];
        }
        rowtop = s_above + s2;
    }

    float wtot = wave_reduce_sum(rowtop);                    // v_wmma_f32_16x16x4_f32 path
    const int wid = t >> 5;
    if ((t & 31) == 0) wpart[wid] = wtot;
    __syncthreads();
    if (t == 0) out[0] = (wpart[0] + wpart[1]) / (float)TOTALK;
}

// ---------------- host entry ----------------
extern "C" void kernel_launch(void* const* d_in, const int* in_sizes, int n_in,
                              void* d_out, int out_size, void* d_ws, size_t ws_size,
                              hipStream_t stream) {
    const float* pred = (const float*)d_in[0];
    const float* gts  = (const float*)d_in[1];
    uint32_t*    ws   = (uint32_t*)d_ws;
    float*       out  = (float*)d_out;

    btk_init<<<(W_END + 255) / 256, 256, 0, stream>>>(ws);

    dim3 grid(SLICES, BATCH);             // 16 x 64 blocks, 8 double-buffered tiles each
    btk_hist1<<<grid, 256, 0, stream>>>(pred, gts, ws);
    btk_select<<<1, 64, 0, stream>>>(ws);
    btk_hist2<<<grid, 256, 0, stream>>>(pred, gts, ws);
    btk_finalize<<<1, 64, 0, stream>>>(ws, out);
}